// MyLoss_33758442946604
// MI455X (gfx1250) — compile-verified
//
#include <hip/hip_runtime.h>

// Problem geometry (fixed by the reference): B=8, S=300, C=4, H=W=32.
// output/convdata: [B,S,4,32,32] fp32 ; visweight: [B,S,32,32] fp32.
// HW = 1024 floats per channel; per (b,s): 4 channel planes of 1024 floats.
#define NBS        2400      // B*S
#define HW4        256       // 1024/4 (float4 units per plane)
#define TOTAL4     (NBS * HW4)   // 614400 float4-granularity work items
#define NBLOCKS    1024
#define NTHREADS   256       // 8 waves of 32 on CDNA5

typedef __attribute__((ext_vector_type(2))) float v2f;
typedef __attribute__((ext_vector_type(8))) float v8f;

// Wave32 sum via V_WMMA_F32_16X16X4_F32.
// A (16x4 f32, 2 VGPRs/lane): lane L<16 holds A[L][0],A[L][1]; lane L>=16 holds
// A[L-16][2],A[L-16][3]. Setting a={acc,0} gives row m = {acc_m,0,acc_{m+16},0}.
// With B = all ones, D[m][n] = acc_m + acc_{m+16} for every n.
// Summing the 8 D VGPRs per lane yields sum(rows 0..7) on lanes 0..15 and
// sum(rows 8..15) on lanes 16..31; two shuffles give the full wave sum.
__device__ __forceinline__ float wave_reduce_wmma(float acc) {
    v2f a; a[0] = acc;  a[1] = 0.0f;
    v2f b; b[0] = 1.0f; b[1] = 1.0f;
    v8f c = {};
    v8f d = __builtin_amdgcn_wmma_f32_16x16x4_f32(
        /*neg_a=*/false, a, /*neg_b=*/false, b,
        /*c_mod=*/(short)0, c, /*reuse_a=*/false, /*reuse_b=*/false);
    float t = ((d[0] + d[1]) + (d[2] + d[3])) + ((d[4] + d[5]) + (d[6] + d[7]));
    return __shfl(t, 0, 32) + __shfl(t, 16, 32);
}

__device__ __forceinline__ float block_reduce(float acc) {
    __shared__ float smem[NTHREADS / 32];
    const int lane = threadIdx.x & 31;
    const int wave = threadIdx.x >> 5;
    float wsum = wave_reduce_wmma(acc);   // EXEC all-1s here (converged)
    if (lane == 0) smem[wave] = wsum;
    __syncthreads();
    float s = 0.0f;
    if (threadIdx.x == 0) {
#pragma unroll
        for (int i = 0; i < NTHREADS / 32; ++i) s += smem[i];
    }
    return s;   // valid on thread 0 only
}

// Pass 1: fused streaming pass. float4 (b128) loads, fully coalesced.
__global__ void __launch_bounds__(NTHREADS)
wloss_partial_kernel(const float* __restrict__ outp,
                     const float* __restrict__ conv,
                     const float* __restrict__ vis,
                     float* __restrict__ partials) {
    const float4* __restrict__ o4 = (const float4*)outp;
    const float4* __restrict__ c4 = (const float4*)conv;
    const float4* __restrict__ w4 = (const float4*)vis;

    float acc = 0.0f;
    const int stride = gridDim.x * blockDim.x;
    for (int idx = blockIdx.x * blockDim.x + threadIdx.x; idx < TOTAL4; idx += stride) {
        const int bs = idx >> 8;          // (b,s) plane group
        const int q  = idx & (HW4 - 1);   // float4 position within 32x32 plane
        const float4 w = w4[idx];
        const size_t base = (size_t)bs * (4 * HW4) + (size_t)q;  // float4 units
#pragma unroll
        for (int ch = 0; ch < 4; ++ch) {
            const float4 o = o4[base + (size_t)ch * HW4];
            const float4 g = c4[base + (size_t)ch * HW4];
            const float dx = g.x - o.x;
            const float dy = g.y - o.y;
            const float dz = g.z - o.z;
            const float dw = g.w - o.w;
            acc = fmaf(w.x, dx * dx, acc);
            acc = fmaf(w.y, dy * dy, acc);
            acc = fmaf(w.z, dz * dz, acc);
            acc = fmaf(w.w, dw * dw, acc);
        }
    }

    const float bsum = block_reduce(acc);
    if (threadIdx.x == 0) partials[blockIdx.x] = bsum;
}

// Pass 2: single block folds NBLOCKS partials, applies /1024.
__global__ void __launch_bounds__(NTHREADS)
wloss_final_kernel(const float* __restrict__ partials,
                   float* __restrict__ out) {
    float acc = 0.0f;
    // NBLOCKS is a multiple of NTHREADS: every lane does exactly 4 iters, no divergence.
    for (int i = threadIdx.x; i < NBLOCKS; i += NTHREADS) acc += partials[i];
    const float s = block_reduce(acc);
    if (threadIdx.x == 0) out[0] = s * (1.0f / 1024.0f);
}

extern "C" void kernel_launch(void* const* d_in, const int* in_sizes, int n_in,
                              void* d_out, int out_size, void* d_ws, size_t ws_size,
                              hipStream_t stream) {
    const float* outp = (const float*)d_in[0];  // "output"   [8,300,4,32,32]
    const float* conv = (const float*)d_in[1];  // "convdata" [8,300,4,32,32]
    const float* vis  = (const float*)d_in[2];  // "visweight"[8,300,32,32]
    float* partials = (float*)d_ws;             // NBLOCKS floats of scratch
    float* out = (float*)d_out;

    wloss_partial_kernel<<<NBLOCKS, NTHREADS, 0, stream>>>(outp, conv, vis, partials);
    wloss_final_kernel<<<1, NTHREADS, 0, stream>>>(partials, out);
}